// TcdLoss_19250043421333
// MI455X (gfx1250) — compile-verified
//
#include <hip/hip_runtime.h>
#include <hip/hip_bf16.h>
#include <stdint.h>

#define B_   64
#define NG_  512
#define NP_  2048
#define C_   80
#define THREADS 256

// ---- CDNA5 async global -> LDS copy (ASYNCcnt path), per ISA 15.18.3 ----
__device__ __forceinline__ uint32_t lds_addr_of(const void* p) {
  // generic LDS pointer = {SHARED_BASE, addr[31:0]}; low 32 bits are the LDS byte address
  return (uint32_t)(uintptr_t)p;
}
__device__ __forceinline__ void async_copy_b128(uint32_t lds_addr, const void* gbase, uint32_t byte_off) {
  asm volatile("global_load_async_to_lds_b128 %0, %1, %2"
               : : "v"(lds_addr), "v"(byte_off), "s"(gbase) : "memory");
}
__device__ __forceinline__ void async_copy_b32(uint32_t lds_addr, const void* gbase, uint32_t byte_off) {
  asm volatile("global_load_async_to_lds_b32 %0, %1, %2"
               : : "v"(lds_addr), "v"(byte_off), "s"(gbase) : "memory");
}
__device__ __forceinline__ void wait_async_all() {
  asm volatile("s_wait_asynccnt 0" : : : "memory");
}

// IoU with +1 convention; MUST be the identical expression in both passes so that
// the pass-2 equality test (iou == rowmax) is bit-exact.
__device__ __forceinline__ float iou_pair(float4 g, float ga, float4 p, float pa) {
  float xx1 = fmaxf(g.x, p.x);
  float yy1 = fmaxf(g.y, p.y);
  float xx2 = fminf(g.z, p.z);
  float yy2 = fminf(g.w, p.w);
  float w = fmaxf(0.0f, xx2 - xx1 + 1.0f);
  float h = fmaxf(0.0f, yy2 - yy1 + 1.0f);
  float inter = w * h;
  return inter / (ga + pa - inter);
}

__global__ __launch_bounds__(THREADS) void tcd_batch_kernel(
    const float* __restrict__ gt_boxes,     // [B,NG,4]
    const float* __restrict__ pred_boxes,   // [B,NP,4]
    const float* __restrict__ scores,       // [B,NP]
    const int*   __restrict__ gt_classes,   // [B,NG]
    const int*   __restrict__ pred_classes, // [B,NP]
    float* __restrict__ wsSum,              // [B]
    float* __restrict__ wsCnt)              // [B]
{
  __shared__ float4 sPredBox[NP_];    // 32 KB
  __shared__ float4 sGtBox[NG_];      //  8 KB
  __shared__ int    sPredCls[NP_];    //  8 KB
  __shared__ int    sGtCls[NG_];      //  2 KB
  __shared__ float  sScore[NP_];      //  8 KB
  __shared__ float  sPredArea[NP_];   //  8 KB
  __shared__ float  sGtArea[NG_];     //  2 KB
  __shared__ float  sRowMax[NG_];     //  2 KB
  __shared__ float  sTerm[NP_];       //  8 KB
  __shared__ int    sTermK[NP_];      //  8 KB
  // per-class CSR buckets (counting sort by class)
  __shared__ int    sPredCnt[C_];
  __shared__ int    sPredStart[C_];
  __shared__ int    sPredCur[C_];
  __shared__ int    sPredIdx[NP_];    //  8 KB
  __shared__ int    sGtCnt[C_];
  __shared__ int    sGtStart[C_];
  __shared__ int    sGtCur[C_];
  __shared__ int    sGtIdx[NG_];      //  2 KB
  __shared__ float  sRedV[THREADS];
  __shared__ float  sRedC[THREADS];

  const int tid = threadIdx.x;
  const int b   = blockIdx.x;

  const float* gB = gt_boxes     + (size_t)b * NG_ * 4;
  const float* pB = pred_boxes   + (size_t)b * NP_ * 4;
  const float* sc = scores       + (size_t)b * NP_;
  const int*   gC = gt_classes   + (size_t)b * NG_;
  const int*   pC = pred_classes + (size_t)b * NP_;

  // ---- Stage all per-batch tensors into LDS via CDNA5 async copies ----
  for (int k = tid; k < NP_; k += THREADS) {
    async_copy_b128(lds_addr_of(&sPredBox[k]), pB, (uint32_t)k * 16u);
    async_copy_b32 (lds_addr_of(&sPredCls[k]), pC, (uint32_t)k * 4u);
    async_copy_b32 (lds_addr_of(&sScore[k]),   sc, (uint32_t)k * 4u);
  }
  for (int k = tid; k < NG_; k += THREADS) {
    async_copy_b128(lds_addr_of(&sGtBox[k]), gB, (uint32_t)k * 16u);
    async_copy_b32 (lds_addr_of(&sGtCls[k]), gC, (uint32_t)k * 4u);
  }
  if (tid < C_) { sPredCnt[tid] = 0; sGtCnt[tid] = 0; }
  wait_async_all();
  __syncthreads();

  // ---- Areas + per-class counts ----
  for (int k = tid; k < NP_; k += THREADS) {
    float4 p = sPredBox[k];
    sPredArea[k] = (p.z - p.x + 1.0f) * (p.w - p.y + 1.0f);
    atomicAdd(&sPredCnt[sPredCls[k]], 1);
  }
  for (int k = tid; k < NG_; k += THREADS) {
    float4 g = sGtBox[k];
    sGtArea[k] = (g.z - g.x + 1.0f) * (g.w - g.y + 1.0f);
    atomicAdd(&sGtCnt[sGtCls[k]], 1);
  }
  __syncthreads();

  // ---- Prefix sums (tiny, serial on thread 0) + cursors ----
  if (tid == 0) {
    int accP = 0, accG = 0;
    for (int c = 0; c < C_; ++c) {
      sPredStart[c] = accP; sPredCur[c] = accP; accP += sPredCnt[c];
      sGtStart[c]   = accG; sGtCur[c]   = accG; accG += sGtCnt[c];
    }
  }
  __syncthreads();

  // ---- Scatter indices into class buckets (order within bucket is irrelevant:
  //      all consumers are max / OR / recomputed-equality, order-independent) ----
  for (int k = tid; k < NP_; k += THREADS) {
    int slot = atomicAdd(&sPredCur[sPredCls[k]], 1);
    sPredIdx[slot] = k;
  }
  for (int k = tid; k < NG_; k += THREADS) {
    int slot = atomicAdd(&sGtCur[sGtCls[k]], 1);
    sGtIdx[slot] = k;
  }
  __syncthreads();

  // ---- Pass 1: per-GT row max over same-class preds only (~NP/C candidates) ----
  for (int i = tid; i < NG_; i += THREADS) {
    float4 g  = sGtBox[i];
    float  ga = sGtArea[i];
    int    gc = sGtCls[i];
    int start = sPredStart[gc];
    int cnt   = sPredCnt[gc];
    float best = 0.0f;  // masked IoU is 0 for class mismatch
    for (int t = 0; t < cnt; ++t) {
      int j = sPredIdx[start + t];
      float iou = iou_pair(g, ga, sPredBox[j], sPredArea[j]);
      best = fmaxf(best, iou);
    }
    sRowMax[i] = best;
  }
  __syncthreads();

  // ---- Pass 2: per-pred matched flag over same-class GTs only (~NG/C candidates) ----
  for (int j = tid; j < NP_; j += THREADS) {
    float4 p  = sPredBox[j];
    float  pa = sPredArea[j];
    int    pc = sPredCls[j];
    int start = sGtStart[pc];
    int cnt   = sGtCnt[pc];
    bool matched = false;
    for (int t = 0; t < cnt; ++t) {
      int i = sGtIdx[start + t];
      float iou = iou_pair(sGtBox[i], sGtArea[i], p, pa);
      matched = matched || ((iou > 0.5f) && (iou == sRowMax[i]));
    }
    float s  = sScore[j];
    float tt = tanhf(s);
    bool  hi = (s >= 0.5f);
    float term;
    int   k;
    if (matched) { term = hi ? (s * tt) : (s * (1.0f - tt));                   k = hi ? 0 : 1; }
    else         { term = hi ? ((1.0f - s) * tt) : ((1.0f - s) * (1.0f - tt)); k = hi ? 2 : 3; }
    sTerm[j]  = term;
    sTermK[j] = k;
  }
  __syncthreads();

  // ---- Pass 3: deterministic (fixed-order) per-class accumulation + log1p ----
  float val = 0.0f, cnt = 0.0f;
  if (tid < C_) {
    float nAC = 0.0f, nAN = 0.0f, nIC = 0.0f, nIN = 0.0f;
    for (int j = 0; j < NP_; ++j) {        // fixed order: replay-deterministic sums
      if (sPredCls[j] == tid) {
        float t = sTerm[j];
        int   k = sTermK[j];
        nAC += (k == 0) ? t : 0.0f;
        nAN += (k == 1) ? t : 0.0f;
        nIC += (k == 2) ? t : 0.0f;
        nIN += (k == 3) ? t : 0.0f;
      }
    }
    float numr  = nAN + nIC;
    float denom = nAC + nIN;
    bool  use   = (sGtCnt[tid] > 0) && (sPredCnt[tid] > 0) && (denom > 0.0f);
    float det   = log1pf(numr / (denom > 0.0f ? denom : 1.0f));
    val = use ? det : 0.0f;
    cnt = use ? 1.0f : 0.0f;
  }

  // ---- Fixed-order block tree reduction (deterministic) ----
  sRedV[tid] = val;
  sRedC[tid] = cnt;
  __syncthreads();
  for (int off = THREADS / 2; off > 0; off >>= 1) {
    if (tid < off) {
      sRedV[tid] += sRedV[tid + off];
      sRedC[tid] += sRedC[tid + off];
    }
    __syncthreads();
  }
  if (tid == 0) {
    wsSum[b] = sRedV[0];
    wsCnt[b] = sRedC[0];
  }
}

__global__ void tcd_finalize_kernel(const float* __restrict__ wsSum,
                                    const float* __restrict__ wsCnt,
                                    float* __restrict__ out)
{
  if (blockIdx.x == 0 && threadIdx.x == 0) {
    float s = 0.0f, c = 0.0f;
    for (int b = 0; b < B_; ++b) { s += wsSum[b]; c += wsCnt[b]; }
    out[0] = s / fmaxf(c, 1.0f);
  }
}

extern "C" void kernel_launch(void* const* d_in, const int* in_sizes, int n_in,
                              void* d_out, int out_size, void* d_ws, size_t ws_size,
                              hipStream_t stream) {
  const float* gt_boxes     = (const float*)d_in[0];
  const float* pred_boxes   = (const float*)d_in[1];
  const float* scores       = (const float*)d_in[2];
  const int*   gt_classes   = (const int*)d_in[3];
  const int*   pred_classes = (const int*)d_in[4];
  float* wsSum = (float*)d_ws;
  float* wsCnt = wsSum + B_;
  float* out   = (float*)d_out;

  tcd_batch_kernel<<<B_, THREADS, 0, stream>>>(gt_boxes, pred_boxes, scores,
                                               gt_classes, pred_classes, wsSum, wsCnt);
  tcd_finalize_kernel<<<1, 32, 0, stream>>>(wsSum, wsCnt, out);
}